// ResidualVectorQuantize2d_52312701665800
// MI455X (gfx1250) — compile-verified
//
#include <hip/hip_runtime.h>
#include <hip/hip_bf16.h>

// Problem constants (from reference)
#define B_      8
#define C_      512
#define H_      32
#define W_      32
#define HW_     (H_ * W_)          // 1024
#define S_      (B_ * HW_)         // 8192 spatial positions
#define K_      8192               // codebook entries
#define D_      16                 // code dim
#define NCB_    4                  // codebooks
#define NCHUNK_ 8                  // k-range split for the argmin kernel

typedef float v2f __attribute__((ext_vector_type(2)));
typedef float v8f __attribute__((ext_vector_type(8)));

// ---------------------------------------------------------------------------
// init: copy z -> residual (must not mutate input), zero z_q_total and SSE acc
// ---------------------------------------------------------------------------
__global__ void rvq_init_kernel(const float* __restrict__ z,
                                float* __restrict__ resid,
                                float* __restrict__ zqt,
                                float* __restrict__ sse) {
    int i = blockIdx.x * blockDim.x + threadIdx.x;
    if (i < S_ * C_) resid[i] = z[i];
    if (i < S_ * C_ / 2) zqt[i] = 0.0f;   // z_q_total: B*C*H*W = 2,097,152 floats
    if (i == 0) *sse = 0.0f;
}

// ---------------------------------------------------------------------------
// codebook squared norms for all 4 codebooks:  cbn[n*K + k] = sum_d cb^2
// ---------------------------------------------------------------------------
__global__ void rvq_cbnorm_kernel(const float* __restrict__ cb,
                                  float* __restrict__ cbn) {
    int t = blockIdx.x * blockDim.x + threadIdx.x;   // 0 .. NCB_*K_-1
    if (t >= NCB_ * K_) return;
    const float* r = cb + (size_t)t * D_;
    float ss = 0.0f;
#pragma unroll
    for (int d = 0; d < D_; ++d) ss += r[d] * r[d];
    cbn[t] = ss;
}

// ---------------------------------------------------------------------------
// weight-norm for the input conv: w_in[d][c] = g[d] * v[d][c] / ||v[d,:]||
// ---------------------------------------------------------------------------
__global__ void rvq_win_kernel(const float* __restrict__ v,
                               const float* __restrict__ g,
                               float* __restrict__ w) {
    __shared__ float red[256];
    __shared__ float scale_s;
    int d = blockIdx.x;
    int t = threadIdx.x;
    float ss = 0.0f;
    for (int c = t; c < C_; c += 256) {
        float x = v[d * C_ + c];
        ss += x * x;
    }
    red[t] = ss;
    __syncthreads();
    for (int o = 128; o > 0; o >>= 1) {
        if (t < o) red[t] += red[t + o];
        __syncthreads();
    }
    if (t == 0) scale_s = g[d] / sqrtf(red[0]);
    __syncthreads();
    float sc = scale_s;
    for (int c = t; c < C_; c += 256)
        w[d * C_ + c] = v[d * C_ + c] * sc;
}

// ---------------------------------------------------------------------------
// weight-norm for the output conv: w_out[c][d] = g[c] * v[c][d] / ||v[c,:]||
// ---------------------------------------------------------------------------
__global__ void rvq_wout_kernel(const float* __restrict__ v,
                                const float* __restrict__ g,
                                float* __restrict__ w) {
    int c = blockIdx.x * blockDim.x + threadIdx.x;
    if (c >= C_) return;
    const float* r = v + c * D_;
    float ss = 0.0f;
#pragma unroll
    for (int d = 0; d < D_; ++d) ss += r[d] * r[d];
    float sc = g[c] / sqrtf(ss);
    float* o = w + c * D_;
#pragma unroll
    for (int d = 0; d < D_; ++d) o[d] = r[d] * sc;
}

// ---------------------------------------------------------------------------
// z_e = residual @ w_in^T + b_in.  Two threads per spatial position, each
// producing 8 of the 16 outputs (doubles wave count on the critical path).
// ---------------------------------------------------------------------------
__global__ void rvq_ze_kernel(const float* __restrict__ resid,
                              const float* __restrict__ w_in,   // [16][512]
                              const float* __restrict__ b_in,   // [16]
                              float* __restrict__ ze) {
    __shared__ float wl[D_ * C_];                 // 32 KB
    int t = threadIdx.x;
    for (int k = t; k < D_ * C_; k += 256) wl[k] = w_in[k];
    __syncthreads();

    int g  = blockIdx.x * 256 + t;    // 16384 threads total
    int s  = g >> 1;
    int dh = (g & 1) * 8;             // d range [dh, dh+8)
    int b  = s >> 10;                 // / HW_
    int hw = s & (HW_ - 1);
    const float* rp = resid + (size_t)b * C_ * HW_ + hw;

    float acc[8];
#pragma unroll
    for (int d = 0; d < 8; ++d) acc[d] = b_in[dh + d];

    for (int c = 0; c < C_; ++c) {
        float r = rp[(size_t)c * HW_];
#pragma unroll
        for (int d = 0; d < 8; ++d)
            acc[d] = fmaf(r, wl[(dh + d) * C_ + c], acc[d]);
    }
    float4* o = (float4*)(ze + (size_t)s * D_ + dh);
    o[0] = make_float4(acc[0], acc[1], acc[2], acc[3]);
    o[1] = make_float4(acc[4], acc[5], acc[6], acc[7]);
}

// ---------------------------------------------------------------------------
// Fused distance + partial argmin via V_WMMA_F32_16X16X4_F32.
// Each wave owns a 16-row tile of ze (A fragments register-resident for all 4
// K-steps) and scans one 1/NCHUNK_ slice of the codebook, tracking a per-lane
// running min of  score(k) = ||cb_k||^2 - 2 * <ze_s, cb_k>.
// Blocks are chunk-major: all 8 waves of a block share a k-chunk and take
// consecutive s-tiles, so they stream identical codebook bytes (L0-friendly).
// A 16x4 f32 fragment: lanes 0-15 -> M=lane, v0=K0,v1=K1; lanes 16-31 -> K2,K3.
// B 4x16 f32 fragment mirrors with N across lanes. Both = one b64 load.
// D tile: acc[j] lanes 0-15 = row j, lanes 16-31 = row j+8, col = lane&15.
// ---------------------------------------------------------------------------
__global__ void rvq_dist_argmin_kernel(const float* __restrict__ ze,
                                       const float* __restrict__ cb,    // [K][16]
                                       const float* __restrict__ cbn,   // [K]
                                       float* __restrict__ pminv,       // [NCHUNK][S]
                                       int*   __restrict__ pmini) {     // [NCHUNK][S]
    const int lane  = threadIdx.x & 31;
    const int wave  = threadIdx.x >> 5;
    const int chunk = blockIdx.x >> 6;               // 64 blocks per chunk
    const int stile = ((blockIdx.x & 63) << 3) + wave;
    const int s0    = stile * 16;
    const int half  = lane >> 4;                     // 0 or 1
    const int l15   = lane & 15;

    // A fragments for the 4 K-steps (d = 4t .. 4t+3), register-resident
    v2f afrag[4];
    {
        const float* ap = ze + (size_t)(s0 + l15) * D_ + half * 2;
#pragma unroll
        for (int t = 0; t < 4; ++t)
            afrag[t] = *(const v2f*)(ap + 4 * t);
    }

    float minv[8];
    int   mini[8];
#pragma unroll
    for (int j = 0; j < 8; ++j) { minv[j] = 3.4e38f; mini[j] = 0; }

    const float* bbase = cb + (size_t)l15 * D_ + half * 2;

    const int kt_beg = chunk * (K_ / 16 / NCHUNK_);   // 64 tiles per chunk
    const int kt_end = kt_beg + (K_ / 16 / NCHUNK_);
#pragma unroll 2
    for (int kt = kt_beg; kt < kt_end; ++kt) {
        const int k0 = kt * 16;
        const float* bp = bbase + (size_t)k0 * D_;

        // prefetch a future codebook tile (emits global_prefetch_b8)
        __builtin_prefetch(bp + 64 * D_, 0, 1);

        v8f acc = {};
#pragma unroll
        for (int t = 0; t < 4; ++t) {
            v2f bfrag = *(const v2f*)(bp + 4 * t);
            acc = __builtin_amdgcn_wmma_f32_16x16x4_f32(
                false, afrag[t], false, bfrag, (short)0, acc, false, false);
        }

        const int   kc = k0 + l15;
        const float cv = cbn[kc];
#pragma unroll
        for (int j = 0; j < 8; ++j) {
            float score = fmaf(-2.0f, acc[j], cv);
            bool lt = score < minv[j];      // strict: earlier k wins ties
            minv[j] = lt ? score : minv[j];
            mini[j] = lt ? kc : mini[j];
        }
    }

    // reduce the 16 candidate columns (lanes) within each half-wave
    // (branch-free tie-break: bitwise ops avoid cmpx/execz divergence)
#pragma unroll
    for (int j = 0; j < 8; ++j) {
        float v = minv[j];
        int   ii = mini[j];
#pragma unroll
        for (int m = 8; m >= 1; m >>= 1) {
            float ov = __shfl_xor(v, m, 32);
            int   oi = __shfl_xor(ii, m, 32);
            int better = (int)(ov < v) | ((int)(ov == v) & (int)(oi < ii));
            v  = better ? ov : v;
            ii = better ? oi : ii;
        }
        if (l15 == 0) {
            int s = s0 + j + half * 8;
            pminv[chunk * S_ + s] = v;
            pmini[chunk * S_ + s] = ii;
        }
    }
}

// ---------------------------------------------------------------------------
// combine partial argmins, gather z_q = cb[idx], accumulate SSE, emit codes
// ---------------------------------------------------------------------------
__global__ void rvq_gather_loss_kernel(const float* __restrict__ ze,
                                       const float* __restrict__ cb,
                                       const float* __restrict__ pminv,
                                       const int*   __restrict__ pmini,
                                       int*   __restrict__ codes,
                                       float* __restrict__ zq,
                                       float* __restrict__ sse) {
    __shared__ float red[256];
    int t = threadIdx.x;
    int s = blockIdx.x * 256 + t;

    // combine the NCHUNK_ partials; ascending chunk + strict < keeps lowest k
    float bv = pminv[s];
    int   bi = pmini[s];
#pragma unroll
    for (int c = 1; c < NCHUNK_; ++c) {
        float v = pminv[c * S_ + s];
        int   i = pmini[c * S_ + s];
        bool lt = v < bv;
        bv = lt ? v : bv;
        bi = lt ? i : bi;
    }
    codes[s] = bi;

    const float* cr = cb + (size_t)bi * D_;
    const float* zr = ze + (size_t)s * D_;
    float*       qr = zq + (size_t)s * D_;
    float acc = 0.0f;
#pragma unroll
    for (int d = 0; d < D_; ++d) {
        float q = cr[d];
        float e = zr[d] - q;
        qr[d] = q;
        acc += e * e;
    }
    red[t] = acc;
    __syncthreads();
    for (int o = 128; o > 0; o >>= 1) {
        if (t < o) red[t] += red[t + o];
        __syncthreads();
    }
    if (t == 0) atomicAdd(sse, red[0]);
}

// ---------------------------------------------------------------------------
// out = z_q @ w_out^T + b_out ; z_q_total += out ; residual -= out
// Two threads per spatial position, each covering 256 of the 512 channels.
// ---------------------------------------------------------------------------
__global__ void rvq_outconv_kernel(const float* __restrict__ zq,
                                   const float* __restrict__ w_out,  // [512][16]
                                   const float* __restrict__ b_out,  // [512]
                                   float* __restrict__ zqt,
                                   float* __restrict__ resid) {
    __shared__ float wl[C_ * D_];                 // 32 KB
    int t = threadIdx.x;
    for (int k = t; k < C_ * D_; k += 256) wl[k] = w_out[k];
    __syncthreads();

    int g  = blockIdx.x * 256 + t;    // 16384 threads total
    int s  = g >> 1;
    int c0 = (g & 1) * (C_ / 2);      // channel range [c0, c0+256)
    int b  = s >> 10;
    int hw = s & (HW_ - 1);

    float q[D_];
    const float4* qr = (const float4*)(zq + (size_t)s * D_);
#pragma unroll
    for (int i = 0; i < 4; ++i) {
        float4 f = qr[i];
        q[4 * i] = f.x; q[4 * i + 1] = f.y; q[4 * i + 2] = f.z; q[4 * i + 3] = f.w;
    }

    float* zp = zqt   + (size_t)b * C_ * HW_ + hw;
    float* rp = resid + (size_t)b * C_ * HW_ + hw;

    for (int c = c0; c < c0 + C_ / 2; ++c) {
        const float* wr = wl + c * D_;
        float dot = b_out[c];
#pragma unroll
        for (int d = 0; d < D_; ++d) dot = fmaf(q[d], wr[d], dot);
        size_t o = (size_t)c * HW_;
        zp[o] += dot;
        rp[o] -= dot;
    }
}

// ---------------------------------------------------------------------------
// final loss scalar: total_loss = (1 + COMMIT_W) * SSE_total / (S*D) / NCB
// ---------------------------------------------------------------------------
__global__ void rvq_loss_final_kernel(const float* __restrict__ sse,
                                      float* __restrict__ loss_out) {
    *loss_out = (*sse) * (1.25f / ((float)(S_ * D_) * (float)NCB_));
}

// ---------------------------------------------------------------------------
extern "C" void kernel_launch(void* const* d_in, const int* in_sizes, int n_in,
                              void* d_out, int out_size, void* d_ws, size_t ws_size,
                              hipStream_t stream) {
    (void)in_sizes; (void)n_in; (void)out_size; (void)ws_size;

    const float* z         = (const float*)d_in[0];   // (8,512,32,32)
    const float* in_v      = (const float*)d_in[1];   // (4,16,512)
    const float* in_g      = (const float*)d_in[2];   // (4,16)
    const float* in_b      = (const float*)d_in[3];   // (4,16)
    const float* out_v     = (const float*)d_in[4];   // (4,512,16)
    const float* out_g     = (const float*)d_in[5];   // (4,512)
    const float* out_b     = (const float*)d_in[6];   // (4,512)
    const float* codebooks = (const float*)d_in[7];   // (4,8192,16)

    float* out      = (float*)d_out;
    float* zqt      = out;                            // 2,097,152 floats
    int*   codes    = (int*)(out + 2097152);          // 4*8192 int32
    float* loss_out = out + 2097152 + NCB_ * S_;      // + 32768

    float* ws    = (float*)d_ws;
    float* resid = ws;                           // 4,194,304
    float* w_in  = resid + (size_t)S_ * C_;      //     8,192
    float* w_out = w_in  + D_ * C_;              //     8,192
    float* ze    = w_out + C_ * D_;              //   131,072
    float* zq    = ze    + (size_t)S_ * D_;      //   131,072
    float* cbn   = zq    + (size_t)S_ * D_;      //    32,768
    float* pminv = cbn   + NCB_ * K_;            //    65,536
    int*   pmini = (int*)(pminv + NCHUNK_ * S_); //    65,536
    float* sse   = (float*)(pmini + NCHUNK_ * S_); //       1

    // init: residual = z, z_q_total = 0, sse = 0
    rvq_init_kernel<<<(S_ * C_ + 255) / 256, 256, 0, stream>>>(z, resid, zqt, sse);

    // all codebook norms up front
    rvq_cbnorm_kernel<<<(NCB_ * K_ + 255) / 256, 256, 0, stream>>>(codebooks, cbn);

    for (int i = 0; i < NCB_; ++i) {
        const float* cb_i = codebooks + (size_t)i * K_ * D_;

        rvq_win_kernel<<<D_, 256, 0, stream>>>(
            in_v + (size_t)i * D_ * C_, in_g + i * D_, w_in);
        rvq_wout_kernel<<<2, 256, 0, stream>>>(
            out_v + (size_t)i * C_ * D_, out_g + i * C_, w_out);

        rvq_ze_kernel<<<2 * S_ / 256, 256, 0, stream>>>(
            resid, w_in, in_b + i * D_, ze);

        // 512 s-tiles x 8 k-chunks, 8 waves/block -> 512 blocks (4096 waves)
        rvq_dist_argmin_kernel<<<(S_ / 16) / 8 * NCHUNK_, 256, 0, stream>>>(
            ze, cb_i, cbn + (size_t)i * K_, pminv, pmini);

        rvq_gather_loss_kernel<<<S_ / 256, 256, 0, stream>>>(
            ze, cb_i, pminv, pmini, codes + (size_t)i * S_, zq, sse);

        rvq_outconv_kernel<<<2 * S_ / 256, 256, 0, stream>>>(
            zq, w_out, out_b + i * C_, zqt, resid);
    }

    rvq_loss_final_kernel<<<1, 1, 0, stream>>>(sse, loss_out);
}